// OrdinalPeakedCELoss_36386962932467
// MI455X (gfx1250) — compile-verified
//
#include <hip/hip_runtime.h>

// ---------------------------------------------------------------------------
// OrdinalPeakedCELoss for MI455X (gfx1250, wave32).
// Stage 1: streaming per-row loss + block partial (WMMA-based wave reduction).
// Stage 2: deterministic single-block final reduction, scaled by 1/B.
// ---------------------------------------------------------------------------

namespace {

constexpr int KC    = 9;    // number of classes
constexpr int BLOCK = 256;  // 8 wave32s per workgroup
constexpr int RPT   = 4;    // rows per thread (4 rows * 36B = 144B, 16B aligned)

typedef __attribute__((ext_vector_type(2))) float v2f;
typedef __attribute__((ext_vector_type(8))) float v8f;

// Sum 32 per-lane floats with one V_WMMA_F32_16X16X4_F32 (ones-vector trick).
// A(16x4) layout (ISA 7.12.2): lane L<16 holds A[L][0..1] in 2 VGPRs,
// lane L>=16 holds A[L-16][2..3]. With B == all-ones and C == 0:
//   D[i][j] = sum_k A[i][k]  (independent of j)
// D layout: lane c<16 VGPR r = D[r][c]; lane c>=16 VGPR r = D[r+8][c-16].
// => lane 0's 8 accumulators sum to lanes {0..7, 16..23};
//    lane 16's 8 accumulators sum to lanes {8..15, 24..31}.
// Caller combines the two half-sums. EXEC must be all ones (wave-uniform call).
__device__ __forceinline__ float wave_halfsum_wmma(float v) {
  v2f a; a[0] = v;    a[1] = 0.0f;
  v2f b; b[0] = 1.0f; b[1] = 1.0f;
  v8f c = {};
  c = __builtin_amdgcn_wmma_f32_16x16x4_f32(false, a, false, b,
                                            (short)0, c, false, false);
  float s = 0.0f;
#pragma unroll
  for (int i = 0; i < 8; ++i) s += c[i];
  return s;
}

// Per-row loss, fully unrolled over K=9, branchless on (k - y).
__device__ __forceinline__ float row_loss(const float* __restrict__ l, int yv) {
  // cdf_weights = widths / 95
  const float CW[KC] = {0.031578947f, 0.073684211f, 0.105263158f,
                        0.105263158f, 0.105263158f, 0.105263158f,
                        0.105263158f, 0.105263158f, 0.263157895f};
  // max for stable softmax
  float m = l[0];
#pragma unroll
  for (int k = 1; k < KC; ++k) m = fmaxf(m, l[k]);

  float e[KC];
  float S = 0.0f;
#pragma unroll
  for (int k = 0; k < KC; ++k) { e[k] = __expf(l[k] - m); S += e[k]; }
  float invS = __builtin_amdgcn_rcpf(S);   // v_rcp_f32 (1 ulp, fine here)

  // NLL: log(S) - (l[y] - m), normalized by 1/ln(9) later
  float ly = 0.0f;
#pragma unroll
  for (int k = 0; k < KC; ++k) ly = (k == yv) ? l[k] : ly;
  float nll = __logf(S) - (ly - m);

  // probs: clamp_min(1e-8) then renormalize (faithful to reference)
  float p[KC];
  float S2 = 0.0f;
#pragma unroll
  for (int k = 0; k < KC; ++k) { p[k] = fmaxf(e[k] * invS, 1e-8f); S2 += p[k]; }
  float inv2 = __builtin_amdgcn_rcpf(S2);

  float py = 0.0f, pl = 0.0f, pr = 0.0f;
  float fm = 0.0f, tail = 0.0f, emd2 = 0.0f, cdf = 0.0f;
#pragma unroll
  for (int k = 0; k < KC; ++k) {
    float pk = p[k] * inv2;
    int d  = k - yv;
    int ad = (d < 0) ? -d : d;
    py = (d ==  0) ? pk : py;
    pl = (d == -1) ? pk : pl;   // 0 if y==0 (k never hits -1)
    pr = (d ==  1) ? pk : pr;   // 0 if y==K-1
    bool far = ad > 1;
    fm = far ? fmaxf(fm, pk) : fm;
    float t = (float)ad - 1.0f;
    tail = far ? fmaf(t * t * t, pk, tail) : tail;
    cdf += pk;
    float diff = cdf - ((d <= 0) ? 1.0f : 0.0f);  // cdf_true[k] = (k <= y)
    emd2 = fmaf(diff * diff, CW[k], emd2);
  }
  float farm  = fmaxf(fm - py + 0.15f, 0.0f);             // DELTA_FAR
  float lpeak = fmaxf(fmaxf(pl, pr) - py + 0.35f, 0.0f);  // PROB_MARGIN
  // ce*1/ln9 + ALPHA*(W_FAR*far + W_TAIL*tail + W_LPEAK*lpeak + W_EMD*emd2)
  return fmaf(0.45511961f, nll,
              0.4f * (7.0f * farm + 9.0f * tail + 12.0f * lpeak + 1.2f * emd2));
}

__global__ __launch_bounds__(BLOCK) void ordinal_loss_partial(
    const float* __restrict__ logits, const int* __restrict__ y,
    float* __restrict__ partials, int B) {
  const int t = blockIdx.x * BLOCK + threadIdx.x;
  const long long row0 = (long long)t * RPT;
  float acc = 0.0f;

  if (row0 + RPT <= B) {
    // 4 rows = 36 floats = 144 bytes, 16B aligned -> 9x global_load_b128
    const float4* g = (const float4*)(logits + row0 * KC);
    float f[KC * RPT];
#pragma unroll
    for (int i = 0; i < KC; ++i) {
      float4 v = g[i];
      f[4 * i + 0] = v.x; f[4 * i + 1] = v.y;
      f[4 * i + 2] = v.z; f[4 * i + 3] = v.w;
    }
    int4 yy = *(const int4*)(y + row0);
    acc  = row_loss(f + 0 * KC, yy.x);
    acc += row_loss(f + 1 * KC, yy.y);
    acc += row_loss(f + 2 * KC, yy.z);
    acc += row_loss(f + 3 * KC, yy.w);
  } else {
    // remainder rows: scalar path
    for (int r = 0; r < RPT; ++r) {
      long long row = row0 + r;
      if (row < B) {
        float f[KC];
#pragma unroll
        for (int k = 0; k < KC; ++k) f[k] = logits[row * KC + k];
        acc += row_loss(f, y[row]);
      }
    }
  }

  // Block reduction: 256 -> 32 via LDS, 32 -> 2 via WMMA, 2 -> 1 via LDS.
  __shared__ float sdata[BLOCK + 2];
  sdata[threadIdx.x] = acc;
  __syncthreads();
  if (threadIdx.x < 32) {            // wave-uniform: wave 0 has EXEC all ones
    float v = 0.0f;
#pragma unroll
    for (int i = 0; i < BLOCK / 32; ++i) v += sdata[threadIdx.x + i * 32];
    float s = wave_halfsum_wmma(v);
    if (threadIdx.x == 0)  sdata[BLOCK]     = s;  // lanes {0..7,16..23}
    if (threadIdx.x == 16) sdata[BLOCK + 1] = s;  // lanes {8..15,24..31}
  }
  __syncthreads();
  if (threadIdx.x == 0) partials[blockIdx.x] = sdata[BLOCK] + sdata[BLOCK + 1];
}

__global__ __launch_bounds__(BLOCK) void ordinal_loss_finalize(
    const float* __restrict__ partials, int n, float scale,
    float* __restrict__ out) {
  float acc = 0.0f;
  for (int i = threadIdx.x; i < n; i += BLOCK) acc += partials[i];
  __shared__ float sdata[BLOCK + 2];
  sdata[threadIdx.x] = acc;
  __syncthreads();
  if (threadIdx.x < 32) {
    float v = 0.0f;
#pragma unroll
    for (int i = 0; i < BLOCK / 32; ++i) v += sdata[threadIdx.x + i * 32];
    float s = wave_halfsum_wmma(v);
    if (threadIdx.x == 0)  sdata[BLOCK]     = s;
    if (threadIdx.x == 16) sdata[BLOCK + 1] = s;
  }
  __syncthreads();
  if (threadIdx.x == 0) out[0] = (sdata[BLOCK] + sdata[BLOCK + 1]) * scale;
}

}  // namespace

extern "C" void kernel_launch(void* const* d_in, const int* in_sizes, int n_in,
                              void* d_out, int out_size, void* d_ws,
                              size_t ws_size, hipStream_t stream) {
  const float* logits = (const float*)d_in[0];
  const int*   y      = (const int*)d_in[1];
  float* out      = (float*)d_out;
  float* partials = (float*)d_ws;

  const int B = in_sizes[1];                 // rows (y has B elements)
  const int rowsPerBlock = BLOCK * RPT;      // 1024
  const int nBlocks = (B + rowsPerBlock - 1) / rowsPerBlock;  // 4096 for B=4M

  ordinal_loss_partial<<<nBlocks, BLOCK, 0, stream>>>(logits, y, partials, B);

  const float scale = (float)(1.0 / (double)B);
  ordinal_loss_finalize<<<1, BLOCK, 0, stream>>>(partials, nBlocks, scale, out);
}